// GCN_11699490914960
// MI455X (gfx1250) — compile-verified
//
#include <hip/hip_runtime.h>
#include <cmath>

typedef __attribute__((ext_vector_type(2))) float v2f;
typedef __attribute__((ext_vector_type(8))) float v8f;

#define N_IN_F   165
#define N_HID_F  128
#define N_EMB_F  2
#define K_PAD    168            // 165 rounded up to multiple of 4
#define BSTRIDE  170            // LDS row stride (even -> 8B aligned)

// ---------------------------------------------------------------- utilities
__global__ void k_fill(float* __restrict__ p, float v, int n) {
    int i = blockIdx.x * blockDim.x + threadIdx.x;
    if (i < n) p[i] = v;
}

__global__ void k_count_deg(const int* __restrict__ dst, float* __restrict__ deg, int nE) {
    int e = blockIdx.x * blockDim.x + threadIdx.x;
    if (e < nE) atomicAdd(&deg[dst[e]], 1.0f);
}

__global__ void k_rsqrt_inplace(float* __restrict__ p, int n) {
    int i = blockIdx.x * blockDim.x + threadIdx.x;
    if (i < n) p[i] = rsqrtf(p[i]);
}

// ------------------------------------------------- GEMM1: H1 = X @ W1 (fp32 WMMA)
// One wave computes a 16(M) x 128(N) strip with V_WMMA_F32_16X16X4_F32.
// W1^T staged in LDS, zero-padded K 165->168, so the K loop has NO guards.
// __launch_bounds__(256,1): 2 waves/SIMD -> ~512 VGPRs/wave, no accumulator spills.
__global__ void __launch_bounds__(256, 1)
k_gemm1_wmma(const float* __restrict__ X, const float* __restrict__ W,
             float* __restrict__ H, int nNodes) {
    extern __shared__ float bsh[];          // [N_HID_F][BSTRIDE], W^T zero-padded

    // ---- stage W^T into LDS (once per block, shared by all 8 waves)
    for (int idx = threadIdx.x; idx < N_HID_F * K_PAD; idx += blockDim.x) {
        int n = idx / K_PAD;
        int k = idx - n * K_PAD;
        int kc = (k < N_IN_F) ? k : (N_IN_F - 1);          // clamped, always-valid load
        float wv = W[(size_t)kc * N_HID_F + n];
        bsh[n * BSTRIDE + k] = (k < N_IN_F) ? wv : 0.0f;   // value select only
    }
    __syncthreads();

    const int lane   = threadIdx.x & 31;
    const int tileM  = blockIdx.x * (blockDim.x >> 5) + (threadIdx.x >> 5);
    const int nTiles = (nNodes + 15) >> 4;
    if (tileM >= nTiles) return;            // wave-uniform exit (after the sync)

    const int hi = lane >> 4;               // 0 or 1 (K sub-block)
    const int lo = lane & 15;               // M row (A) / N col (B,C)
    int m = tileM * 16 + lo;
    if (m >= nNodes) m = nNodes - 1;        // clamp; stores guarded separately

    v8f zero = {};
    v8f acc[8];
#pragma unroll
    for (int t = 0; t < 8; ++t) acc[t] = zero;

    const float* Xrow  = X + (size_t)m * N_IN_F;
    const float* bbase = bsh + lo * BSTRIDE;

    // ---- main K loop: k = 0 .. 163, fully unguarded; capped unroll to avoid spills
#pragma unroll 2
    for (int k0 = 0; k0 < 164; k0 += 4) {
        const int kA = k0 + 2 * hi;
        v2f a;
        a.x = Xrow[kA];
        a.y = Xrow[kA + 1];
#pragma unroll
        for (int t = 0; t < 8; ++t) {
            v2f b = *(const v2f*)(bbase + (size_t)(t * 16) * BSTRIDE + kA);  // ds_load_b64
            acc[t] = __builtin_amdgcn_wmma_f32_16x16x4_f32(
                false, a, false, b, (short)0, acc[t], false, false);
        }
    }

    // ---- tail: k = 164 (165th column); B rows 165..167 are zero in LDS
    {
        const int kA = 164 + 2 * hi;         // 164 or 166, within padded 168
        float xv = Xrow[164];                // always-valid address
        v2f a;
        a.x = (hi == 0) ? xv : 0.0f;         // v_cndmask on value
        a.y = 0.0f;
#pragma unroll
        for (int t = 0; t < 8; ++t) {
            v2f b = *(const v2f*)(bbase + (size_t)(t * 16) * BSTRIDE + kA);
            acc[t] = __builtin_amdgcn_wmma_f32_16x16x4_f32(
                false, a, false, b, (short)0, acc[t], false, false);
        }
    }

    // ---- store D: VGPR r -> row r (lanes 0-15) / r+8 (lanes 16-31), col = lo
    if (tileM * 16 + 16 <= nNodes) {         // full tile: unguarded stores
        float* Hrow = H + (size_t)(tileM * 16 + 8 * hi) * N_HID_F + lo;
#pragma unroll
        for (int r = 0; r < 8; ++r) {
#pragma unroll
            for (int t = 0; t < 8; ++t)
                Hrow[(size_t)r * N_HID_F + t * 16] = acc[t][r];
        }
    } else {                                 // partial last tile
#pragma unroll
        for (int r = 0; r < 8; ++r) {
            int row = tileM * 16 + r + 8 * hi;
            if (row < nNodes) {
#pragma unroll
                for (int t = 0; t < 8; ++t)
                    H[(size_t)row * N_HID_F + t * 16 + lo] = acc[t][r];
            }
        }
    }
}

// ---------------------------------------- layer-1 aggregation: one wave per edge
__global__ void k_aggregate_hid(const float* __restrict__ H, const int* __restrict__ src,
                                const int* __restrict__ dst, const float* __restrict__ inv,
                                float* __restrict__ out, int nE) {
    const int lane = threadIdx.x & 31;
    const int e = blockIdx.x * (blockDim.x >> 5) + (threadIdx.x >> 5);
    if (e >= nE) return;
    const int s = src[e], d = dst[e];
    const float c = inv[s] * inv[d];
    const float4 v = ((const float4*)(H + (size_t)s * N_HID_F))[lane];
    float* o = out + (size_t)d * N_HID_F + lane * 4;
    atomicAdd(o + 0, v.x * c);
    atomicAdd(o + 1, v.y * c);
    atomicAdd(o + 2, v.z * c);
    atomicAdd(o + 3, v.w * c);
}

// ------------------- fused: h = tanh(agg + h1 * inv^2 + b1), written over h1
__global__ void k_post1(float* __restrict__ H1, const float* __restrict__ agg,
                        const float* __restrict__ inv, const float* __restrict__ b1,
                        int total) {
    int i = blockIdx.x * blockDim.x + threadIdx.x;
    if (i >= total) return;
    int node = i >> 7;            // /128
    int f    = i & 127;
    float is = inv[node];
    float v  = agg[i] + H1[i] * is * is + b1[f];
    H1[i] = tanhf(v);
}

// ------------------------------------------------- GEMM2: H2 = H @ W2 (128 -> 2)
__global__ void k_gemm2(const float* __restrict__ H, const float* __restrict__ W2,
                        float* __restrict__ H2, int nNodes) {
    __shared__ float w2s[N_HID_F * N_EMB_F];  // 256 floats
    int tid = threadIdx.x;
    if (tid < N_HID_F * N_EMB_F) w2s[tid] = W2[tid];
    __syncthreads();
    int n = blockIdx.x * blockDim.x + tid;
    if (n >= nNodes) return;
    const float* h = H + (size_t)n * N_HID_F;
    float a0 = 0.0f, a1 = 0.0f;
#pragma unroll 8
    for (int k = 0; k < N_HID_F; ++k) {
        float hv = h[k];
        a0 += hv * w2s[k * 2 + 0];
        a1 += hv * w2s[k * 2 + 1];
    }
    H2[(size_t)n * 2 + 0] = a0;
    H2[(size_t)n * 2 + 1] = a1;
}

// ------------------------------------ layer-2 aggregation: one thread per edge
__global__ void k_aggregate_emb(const float* __restrict__ H2, const int* __restrict__ src,
                                const int* __restrict__ dst, const float* __restrict__ inv,
                                float* __restrict__ out, int nE) {
    int e = blockIdx.x * blockDim.x + threadIdx.x;
    if (e >= nE) return;
    const int s = src[e], d = dst[e];
    const float c = inv[s] * inv[d];
    atomicAdd(&out[(size_t)d * 2 + 0], H2[(size_t)s * 2 + 0] * c);
    atomicAdd(&out[(size_t)d * 2 + 1], H2[(size_t)s * 2 + 1] * c);
}

// -------- fused: emb = tanh(agg2 + h2*inv^2 + b2); out = sigmoid(emb@Wc + bc)
__global__ void k_final(const float* __restrict__ H2, const float* __restrict__ agg,
                        const float* __restrict__ inv, const float* __restrict__ b2,
                        const float* __restrict__ Wc, const float* __restrict__ bc,
                        float* __restrict__ out, int nNodes) {
    int n = blockIdx.x * blockDim.x + threadIdx.x;
    if (n >= nNodes) return;
    float is = inv[n];
    float i2 = is * is;
    float e0 = tanhf(agg[(size_t)n * 2 + 0] + H2[(size_t)n * 2 + 0] * i2 + b2[0]);
    float e1 = tanhf(agg[(size_t)n * 2 + 1] + H2[(size_t)n * 2 + 1] * i2 + b2[1]);
    float z  = e0 * Wc[0] + e1 * Wc[1] + bc[0];
    out[n] = 1.0f / (1.0f + expf(-z));
}

// ---------------------------------------------------------------- launcher
extern "C" void kernel_launch(void* const* d_in, const int* in_sizes, int n_in,
                              void* d_out, int out_size, void* d_ws, size_t ws_size,
                              hipStream_t stream) {
    const float* x   = (const float*)d_in[0];
    const int*   ei  = (const int*)d_in[1];
    const float* W1  = (const float*)d_in[2];
    const float* b1  = (const float*)d_in[3];
    const float* W2  = (const float*)d_in[4];
    const float* b2  = (const float*)d_in[5];
    const float* Wc  = (const float*)d_in[6];
    const float* bc  = (const float*)d_in[7];
    float* out = (float*)d_out;

    const int nN = in_sizes[0] / N_IN_F;
    const int nE = in_sizes[1] / 2;
    const int* src = ei;
    const int* dst = ei + nE;

    float* ws   = (float*)d_ws;
    float* h1   = ws;                            // nN * 128
    float* agg1 = h1   + (size_t)nN * N_HID_F;   // nN * 128
    float* inv  = agg1 + (size_t)nN * N_HID_F;   // nN   (degree, then rsqrt)
    float* h2   = inv  + (size_t)nN;             // nN * 2
    float* agg2 = h2   + (size_t)nN * 2;         // nN * 2

    const int B   = 256;           // 8 waves per block (wave32)
    const int WPB = B / 32;
    const int hidTotal = nN * N_HID_F;

    // degrees with self-loop, then inv_sqrt (in place)
    k_fill<<<(nN + B - 1) / B, B, 0, stream>>>(inv, 1.0f, nN);
    k_count_deg<<<(nE + B - 1) / B, B, 0, stream>>>(dst, inv, nE);
    k_rsqrt_inplace<<<(nN + B - 1) / B, B, 0, stream>>>(inv, nN);

    // layer 1
    const int nTiles = (nN + 15) / 16;
    const size_t ldsBytes = (size_t)N_HID_F * BSTRIDE * sizeof(float);  // 85 KB
    k_gemm1_wmma<<<(nTiles + WPB - 1) / WPB, B, ldsBytes, stream>>>(x, W1, h1, nN);
    k_fill<<<(hidTotal + B - 1) / B, B, 0, stream>>>(agg1, 0.0f, hidTotal);
    k_aggregate_hid<<<(nE + WPB - 1) / WPB, B, 0, stream>>>(h1, src, dst, inv, agg1, nE);
    k_post1<<<(hidTotal + B - 1) / B, B, 0, stream>>>(h1, agg1, inv, b1, hidTotal);

    // layer 2 + classifier
    k_gemm2<<<(nN + B - 1) / B, B, 0, stream>>>(h1, W2, h2, nN);
    k_fill<<<(nN * 2 + B - 1) / B, B, 0, stream>>>(agg2, 0.0f, nN * 2);
    k_aggregate_emb<<<(nE + B - 1) / B, B, 0, stream>>>(h2, src, dst, inv, agg2, nE);
    k_final<<<(nN + B - 1) / B, B, 0, stream>>>(h2, agg2, inv, b2, Wc, bc, out, nN);
}